// HeteroGNNLayer_1099511628159
// MI455X (gfx1250) — compile-verified
//
#include <hip/hip_runtime.h>

// CDNA5 / gfx1250. D = 128 fixed feature dim.
// out_A = scatter(x_B @ W_msg_ba) + x_A @ W_self_ba
// out_B = scatter(x_A @ W_msg_ab1) + scatter(x_A @ W_msg_ab2) + x_B @ (W_self_ab1 + W_self_ab2)
// Transform-then-scatter: segment_sum(x[src] @ W) == scatter_add((x @ W)[src]).

typedef __attribute__((ext_vector_type(2))) float v2f;
typedef __attribute__((ext_vector_type(8))) float v8f;

#define FDIM 128
#define LDS_STRIDE 132            // transposed W rows padded to 132 floats: b64 loads hit all 64 banks once
#define ROWS_PER_WAVE 32          // two 16-row A fragments share each B fragment (2 WMMA per ds_load_b64)
#define ROWS_PER_BLOCK 256        // 8 waves * 32 rows

// Y[nRows x 128] = X[nRows x 128] @ (W + W2?)   (row-major, fp32, WMMA f32 16x16x4)
__global__ __launch_bounds__(256)
void hgnn_gemm128_f32(const float* __restrict__ X,
                      const float* __restrict__ W,
                      const float* __restrict__ W2,   // nullptr or second weight to add
                      float* __restrict__ Out,
                      int nRows)
{
    // Transposed weight: sWT[col*132 + k]; (k,k+1) contiguous -> one ds_load_b64 per B fragment.
    __shared__ float sWT[FDIM * LDS_STRIDE];

    const int tid  = threadIdx.x;
    const int wave = tid >> 5;
    const int lane = tid & 31;
    const int half = lane >> 4;          // 0: lanes 0-15, 1: lanes 16-31
    const int m    = lane & 15;

    const int rowBase = blockIdx.x * ROWS_PER_BLOCK + wave * ROWS_PER_WAVE;

    // Clamp load rows so EXEC stays all-1s for WMMA on the ragged tail.
    int r0 = rowBase + m;       if (r0 >= nRows) r0 = nRows - 1;
    int r1 = rowBase + 16 + m;  if (r1 >= nRows) r1 = nRows - 1;
    const float* xrow0 = X + (size_t)r0 * FDIM;
    const float* xrow1 = X + (size_t)r1 * FDIM;

    // Warm the A stream (global_prefetch_b8) while we stage W into LDS.
    __builtin_prefetch(xrow0, 0, 0);
    __builtin_prefetch(xrow1, 0, 0);

    // Cooperative transpose-stage of W (+W2) into LDS: coalesced float4 reads,
    // 4 scattered b32 writes (one-time 64KB, amortized over 512KB of ds reads/block).
    {
        const float4* w4  = (const float4*)W;
        const int     nf4 = (FDIM * FDIM) / 4;      // 4096
        if (W2 != nullptr) {
            const float4* w24 = (const float4*)W2;
            for (int i = tid; i < nf4; i += 256) {
                const int k = i >> 5, c = (i & 31) * 4;
                float4 a = w4[i], b = w24[i];
                sWT[(c + 0) * LDS_STRIDE + k] = a.x + b.x;
                sWT[(c + 1) * LDS_STRIDE + k] = a.y + b.y;
                sWT[(c + 2) * LDS_STRIDE + k] = a.z + b.z;
                sWT[(c + 3) * LDS_STRIDE + k] = a.w + b.w;
            }
        } else {
            for (int i = tid; i < nf4; i += 256) {
                const int k = i >> 5, c = (i & 31) * 4;
                float4 a = w4[i];
                sWT[(c + 0) * LDS_STRIDE + k] = a.x;
                sWT[(c + 1) * LDS_STRIDE + k] = a.y;
                sWT[(c + 2) * LDS_STRIDE + k] = a.z;
                sWT[(c + 3) * LDS_STRIDE + k] = a.w;
            }
        }
    }
    __syncthreads();

    v8f acc0[8], acc1[8];
#pragma unroll
    for (int n = 0; n < 8; ++n) {
        acc0[n] = (v8f){0.f,0.f,0.f,0.f,0.f,0.f,0.f,0.f};
        acc1[n] = (v8f){0.f,0.f,0.f,0.f,0.f,0.f,0.f,0.f};
    }

    // K loop: 32 steps of 4. A frag per ISA 16x4 f32 layout:
    //   lanes 0-15 -> K = k0,k0+1 ; lanes 16-31 -> K = k0+2,k0+3 (M = lane%16).
    // B frag mirrors it: single ds_load_b64 from transposed LDS, shared by both row tiles.
#pragma unroll 4
    for (int k0 = 0; k0 < FDIM; k0 += 4) {
        const int k = k0 + 2 * half;
        const v2f a0 = *(const v2f*)(xrow0 + k);     // 8B aligned (k even)
        const v2f a1 = *(const v2f*)(xrow1 + k);
#pragma unroll
        for (int n = 0; n < 8; ++n) {
            const v2f b = *(const v2f*)(sWT + (n * 16 + m) * LDS_STRIDE + k);
            acc0[n] = __builtin_amdgcn_wmma_f32_16x16x4_f32(
                          false, a0, false, b, (short)0, acc0[n], false, false);
            acc1[n] = __builtin_amdgcn_wmma_f32_16x16x4_f32(
                          false, a1, false, b, (short)0, acc1[n], false, false);
        }
    }

    // C/D layout: VGPR i -> lanes 0-15: M=i, N=lane ; lanes 16-31: M=i+8, N=lane-16
#pragma unroll
    for (int i = 0; i < 8; ++i) {
        const int row0 = rowBase + i + 8 * half;
        if (row0 < nRows) {
            float* orow = Out + (size_t)row0 * FDIM;
#pragma unroll
            for (int n = 0; n < 8; ++n)
                orow[n * 16 + m] = acc0[n][i];
        }
        const int row1 = rowBase + 16 + i + 8 * half;
        if (row1 < nRows) {
            float* orow = Out + (size_t)row1 * FDIM;
#pragma unroll
            for (int n = 0; n < 8; ++n)
                orow[n * 16 + m] = acc1[n][i];
        }
    }
}

// One wave per edge: lane loads float4 of Y[src] (32 lanes * 16B = 512B row),
// 4x global_atomic_add_f32 into Out[dst]. Out tables (51MB) stay L2-resident (192MB L2).
__global__ __launch_bounds__(256)
void hgnn_scatter_add(const float* __restrict__ Y,
                      const int* __restrict__ src,
                      const int* __restrict__ dst,
                      float* __restrict__ Out,
                      int nEdges)
{
    const int gid  = blockIdx.x * 256 + threadIdx.x;
    const int e    = gid >> 5;
    const int lane = gid & 31;
    if (e >= nEdges) return;

    const int s = src[e];
    const int d = dst[e];

    const float4 v = ((const float4*)(Y + (size_t)s * FDIM))[lane];
    float* o = Out + (size_t)d * FDIM + lane * 4;
    unsafeAtomicAdd(o + 0, v.x);
    unsafeAtomicAdd(o + 1, v.y);
    unsafeAtomicAdd(o + 2, v.z);
    unsafeAtomicAdd(o + 3, v.w);
}

extern "C" void kernel_launch(void* const* d_in, const int* in_sizes, int n_in,
                              void* d_out, int out_size, void* d_ws, size_t ws_size,
                              hipStream_t stream)
{
    const float* xA        = (const float*)d_in[0];
    const float* xB        = (const float*)d_in[1];
    const float* Wmsg_ab1  = (const float*)d_in[2];
    const float* Wself_ab1 = (const float*)d_in[3];
    const float* Wmsg_ab2  = (const float*)d_in[4];
    const float* Wself_ab2 = (const float*)d_in[5];
    const float* Wmsg_ba   = (const float*)d_in[6];
    const float* Wself_ba  = (const float*)d_in[7];
    const int*   src_ab1   = (const int*)d_in[8];
    const int*   dst_ab1   = (const int*)d_in[9];
    const int*   src_ab2   = (const int*)d_in[10];
    const int*   dst_ab2   = (const int*)d_in[11];
    const int*   src_ba    = (const int*)d_in[12];
    const int*   dst_ba    = (const int*)d_in[13];

    const int NA = in_sizes[0] / FDIM;
    const int NB = in_sizes[1] / FDIM;
    const int E  = in_sizes[8];

    float* outA = (float*)d_out;
    float* outB = outA + (size_t)NA * FDIM;
    float* Y    = (float*)d_ws;              // needs max(NA,NB)*128*4 = 51.2MB; reused serially

    const dim3 blk(256);
    const dim3 gA((NA + ROWS_PER_BLOCK - 1) / ROWS_PER_BLOCK);
    const dim3 gB((NB + ROWS_PER_BLOCK - 1) / ROWS_PER_BLOCK);
    const dim3 gE((unsigned)(((size_t)E * 32 + 255) / 256));

    // Self terms initialize the outputs (overwrites poisoned d_out).
    hgnn_gemm128_f32<<<gA, blk, 0, stream>>>(xA, Wself_ba,  nullptr,   outA, NA);
    hgnn_gemm128_f32<<<gB, blk, 0, stream>>>(xB, Wself_ab1, Wself_ab2, outB, NB);

    // Relation A->B #1
    hgnn_gemm128_f32<<<gA, blk, 0, stream>>>(xA, Wmsg_ab1, nullptr, Y, NA);
    hgnn_scatter_add<<<gE, blk, 0, stream>>>(Y, src_ab1, dst_ab1, outB, E);

    // Relation A->B #2
    hgnn_gemm128_f32<<<gA, blk, 0, stream>>>(xA, Wmsg_ab2, nullptr, Y, NA);
    hgnn_scatter_add<<<gE, blk, 0, stream>>>(Y, src_ab2, dst_ab2, outB, E);

    // Relation B->A
    hgnn_gemm128_f32<<<gB, blk, 0, stream>>>(xB, Wmsg_ba, nullptr, Y, NB);
    hgnn_scatter_add<<<gE, blk, 0, stream>>>(Y, src_ba, dst_ba, outA, E);
}